// DistanceMatrix_42623255446354
// MI455X (gfx1250) — compile-verified
//
#include <hip/hip_runtime.h>
#include <math.h>

// Problem dims from the reference: mask [8, 192, 192] f32.
#define BB 8
#define HH 192
#define WW 192
#define NT (WW / 16)            // 12 tiles of 16 along the width
#define WAVES_PER_BLOCK 4

typedef __attribute__((ext_vector_type(2))) float v2f;
typedef __attribute__((ext_vector_type(8))) float v8f;

// ---------------------------------------------------------------------------
// Kernel 1: vertical L1 pass (faithful to reference cummax formulation),
// writes g2 = d_vert^2 into workspace. One thread per (b, w) column.
// ---------------------------------------------------------------------------
__global__ __launch_bounds__(256) void vert_pass(const float* __restrict__ mask,
                                                 float* __restrict__ g2) {
  int idx = blockIdx.x * blockDim.x + threadIdx.x;
  if (idx >= BB * WW) return;
  int b = idx / WW;
  int w = idx - b * WW;
  const float BIGF = (float)(HH * HH + WW * WW);   // sentinel, matches reference
  const float* mcol = mask + (size_t)b * HH * WW + w;
  float* gcol = g2 + (size_t)b * HH * WW + w;

  // downward scan: up[h] = h - max_{h'<=h, fg} h'   (or h + BIG if none)
  float m_up = -BIGF;
  for (int h = 0; h < HH; ++h) {
    float mv = mcol[(size_t)h * WW];
    float r = (float)h;
    if (mv > 0.5f) m_up = r;            // running max of (fg ? r : -BIG)
    gcol[(size_t)h * WW] = r - m_up;    // stash "up" distance in-place
  }
  // upward scan: down[h] = min_{h'>=h, fg} h' - h   (or BIG - h if none)
  float m_dn = -BIGF;
  for (int h = HH - 1; h >= 0; --h) {
    float mv = mcol[(size_t)h * WW];
    float r = (float)h;
    if (mv > 0.5f) m_dn = -r;           // running max of (fg ? -r : -BIG)
    float up = gcol[(size_t)h * WW];
    float down = -m_dn - r;
    float d = fminf(fminf(up, down), BIGF);
    gcol[(size_t)h * WW] = d * d;       // g2
  }
}

// ---------------------------------------------------------------------------
// Kernel 2: exact 1-D parabolic EDT per row via WMMA f32 16x16x4:
//   A[m,:]  = [ j_m^2, -2*j_m, 1, 0 ]              (j_m = output column)
//   B[:,n]  = [ 1, j'_n, j'_n^2 + g2[j'_n], 0 ]^T  (j'_n = candidate column)
//   D[m,n]  = g2[j'_n] + (j_m - j'_n)^2            (exact: all ints < 2^24)
// then row-min over n (elementwise across tiles + shfl_xor within halves),
// then 2*sigmoid(-0.1*sqrt(D)).
// One wave32 per (b,h) row. EXEC stays all-ones around every WMMA.
// ---------------------------------------------------------------------------
__global__ __launch_bounds__(32 * WAVES_PER_BLOCK)
void edt_rows(const float* __restrict__ g2, float* __restrict__ out) {
  __shared__ float rowmin[WAVES_PER_BLOCK][WW];

  int wave = threadIdx.x >> 5;
  int lane = threadIdx.x & 31;
  int row = blockIdx.x * WAVES_PER_BLOCK + wave;     // 0 .. B*H-1 (exact fit)
  const float* grow = g2 + (size_t)row * WW;

  // CDNA5 global_prefetch_b8: one byte per lane pulls the row's cachelines.
  __builtin_prefetch(grow + lane * 6, 0, 0);

  bool upper = lane >= 16;
  int nl = lane & 15;

  // B tiles (ISA 32-bit 4x16 layout: vgpr0 = rows K0|K2, vgpr1 = rows K1|K3).
  v2f bt[NT];
#pragma unroll
  for (int t = 0; t < NT; ++t) {
    float jp = (float)(t * 16 + nl);
    float g = grow[t * 16 + nl];                 // used by upper half only
    bt[t].x = upper ? (jp * jp + g) : 1.0f;      // B[2,n] : B[0,n]
    bt[t].y = upper ? 0.0f : jp;                 // B[3,n] : B[1,n]
  }

  v8f czero = {};
  for (int o = 0; o < NT; ++o) {
    float j = (float)(o * 16 + nl);
    // A tile (ISA 32-bit 16x4 layout: vgpr0 = cols K0|K2, vgpr1 = cols K1|K3).
    v2f a;
    a.x = upper ? 1.0f : (j * j);                // A[m,2] : A[m,0]
    a.y = upper ? 0.0f : (-2.0f * j);            // A[m,3] : A[m,1]

    v8f acc;
#pragma unroll
    for (int i = 0; i < 8; ++i) acc[i] = 3.0e38f;

#pragma unroll
    for (int t = 0; t < NT; ++t) {
      // D = A x B + 0  ->  v_wmma_f32_16x16x4_f32
      v8f d = __builtin_amdgcn_wmma_f32_16x16x4_f32(
          /*neg_a=*/false, a, /*neg_b=*/false, bt[t],
          /*c_mod=*/(short)0, czero, /*reuse_a=*/false, /*reuse_b=*/false);
#pragma unroll
      for (int i = 0; i < 8; ++i) acc[i] = fminf(acc[i], d[i]);
    }

    // Min over the N dimension: 16 lanes per half (xor masks stay in-half).
#pragma unroll
    for (int i = 0; i < 8; ++i) {
#pragma unroll
      for (int m = 1; m < 16; m <<= 1)
        acc[i] = fminf(acc[i], __shfl_xor(acc[i], m, 32));
    }

    // C/D layout: vgpr r holds M=r (lanes 0-15) and M=r+8 (lanes 16-31).
    if (lane == 0) {
#pragma unroll
      for (int r = 0; r < 8; ++r) rowmin[wave][o * 16 + r] = acc[r];
    }
    if (lane == 16) {
#pragma unroll
      for (int r = 0; r < 8; ++r) rowmin[wave][o * 16 + 8 + r] = acc[r];
    }
  }

  __syncthreads();

  // Coalesced postprocess + store: 2*sigmoid(-0.1*sqrt(dist_sq)).
  for (int k = lane; k < WW; k += 32) {
    float dsq = rowmin[wave][k];
    float v = 2.0f / (1.0f + expf(0.1f * sqrtf(dsq)));
    out[(size_t)row * WW + k] = v;
  }
}

// ---------------------------------------------------------------------------
extern "C" void kernel_launch(void* const* d_in, const int* in_sizes, int n_in,
                              void* d_out, int out_size, void* d_ws, size_t ws_size,
                              hipStream_t stream) {
  (void)in_sizes; (void)n_in; (void)out_size; (void)ws_size;
  const float* mask = (const float*)d_in[0];
  float* out = (float*)d_out;
  float* g2 = (float*)d_ws;                       // B*H*W floats of scratch

  int cols = BB * WW;                             // 1536 columns
  vert_pass<<<(cols + 255) / 256, 256, 0, stream>>>(mask, g2);

  int rows = BB * HH;                             // 1536 rows, 4 waves/block
  edt_rows<<<rows / WAVES_PER_BLOCK, 32 * WAVES_PER_BLOCK, 0, stream>>>(g2, out);
}